// free_NNB_15659450761782
// MI455X (gfx1250) — compile-verified
//
#include <hip/hip_runtime.h>

// ---------------------------------------------------------------------------
// Slater-determinant product kernel for MI455X (gfx1250, wave32).
//   out[b] = det(states_up[idx_up(b)]) * det(states_down[idx_down(b)])
// One wave per batch row (wave32: 32 lanes x 8 flags = 256 sites per half).
// Bandwidth-bound: 256 MB config stream -> ~11 us floor at 23.3 TB/s.
// Gather runs on the matrix pipe: mats = P * states via chained
// V_WMMA_F32_16X16X4_F32 (exact in f32, P is one-hot), skipping empty
// K-chunks via a uniform 64-bit mask walked with scalar ctz. Determinants:
// in-register Gaussian elimination, up/down in parallel across half-waves.
// ---------------------------------------------------------------------------

typedef float v8f __attribute__((ext_vector_type(8)));
typedef float v2f __attribute__((ext_vector_type(2)));

#define NSITES 256          // S
#define WPB 8               // waves per block
#define TSTRIDE 20          // padded LDS tile stride (bank-conflict free, 16B aligned)

__device__ __forceinline__ int bperm_i(int srcLane, int v) {
  return __builtin_amdgcn_ds_bpermute(srcLane << 2, v);
}
__device__ __forceinline__ float bperm_f(int srcLane, float v) {
  return __int_as_float(__builtin_amdgcn_ds_bpermute(srcLane << 2, __float_as_int(v)));
}

// spread 32 bits into even bit positions of a 64-bit word (uniform -> SALU)
__device__ __forceinline__ unsigned long long spread1(unsigned x) {
  unsigned long long v = x;
  v = (v | (v << 16)) & 0x0000FFFF0000FFFFull;
  v = (v | (v << 8))  & 0x00FF00FF00FF00FFull;
  v = (v | (v << 4))  & 0x0F0F0F0F0F0F0F0Full;
  v = (v | (v << 2))  & 0x3333333333333333ull;
  v = (v | (v << 1))  & 0x5555555555555555ull;
  return v;
}

// Accumulate the 16x16 gathered matrix in WMMA C layout.
// A tile (16x4 f32): lane L holds M=L&15, K = 4c + 2*(L>>4) + {0,1}
// B tile (4x16 f32): lane L holds N=L&15, rows K = 4c + 2*(L>>4) + {0,1}
__device__ __forceinline__ v8f gather_wmma(unsigned long long mm, int myIdx,
                                           const float* __restrict__ sArr,
                                           int khalf, int col) {
  v8f acc = {0.f, 0.f, 0.f, 0.f, 0.f, 0.f, 0.f, 0.f};
  // hoist lane-dependent part of the address once
  const float* lanePtr = sArr + khalf * 16 + col;
  while (mm) {
    int c = __builtin_ctzll(mm);        // uniform: scalar ctz + scalar branch
    mm &= mm - 1;
    const int base = c << 2;
    const float* p = lanePtr + (base << 4);   // row (base+khalf), column col
    v2f A, Bv;
    Bv[0] = p[0];                        // offsets 0 / 64B -> foldable DS offsets
    Bv[1] = p[16];
    const int diff = myIdx - (base + khalf);
    A[0] = (diff == 0) ? 1.0f : 0.0f;
    A[1] = (diff == 1) ? 1.0f : 0.0f;
    acc = __builtin_amdgcn_wmma_f32_16x16x4_f32(
        /*neg_a=*/false, A, /*neg_b=*/false, Bv,
        /*c_mod=*/(short)0, acc, /*reuse_a=*/false, /*reuse_b=*/false);
  }
  return acc;
}

__global__ __launch_bounds__(256) void slater_det_kernel(
    const float* __restrict__ cfg,       // [nB, 512]
    const float* __restrict__ statesU,   // [256, 16]
    const float* __restrict__ statesD,   // [256, 16]
    float* __restrict__ out, int nB) {
  __shared__ float sU[NSITES * 16];
  __shared__ float sD[NSITES * 16];
  __shared__ int   idxU[WPB][16];
  __shared__ int   idxD[WPB][16];
  __shared__ float tU[WPB][16 * TSTRIDE];
  __shared__ float tD[WPB][16 * TSTRIDE];

  const int tid = threadIdx.x;

  // cooperative stage of the tiny states matrices into LDS (16 KB each)
  {
    const float4* gu = (const float4*)statesU;
    const float4* gd = (const float4*)statesD;
    float4* lu = (float4*)sU;
    float4* ld = (float4*)sD;
    for (int i = tid; i < (NSITES * 16) / 4; i += 256) { lu[i] = gu[i]; ld[i] = gd[i]; }
  }
  __syncthreads();

  const int wid  = tid >> 5;
  const int lane = tid & 31;
  const long b   = (long)blockIdx.x * WPB + wid;
  if (b >= nB) return;                    // uniform per wave

  // ---- stream this row's occupancy: 4 x global_load_b128 per lane --------
  const float* row = cfg + b * (2 * NSITES);
  float4 u0 = *(const float4*)(row + lane * 8);
  float4 u1 = *(const float4*)(row + lane * 8 + 4);
  float4 d0 = *(const float4*)(row + NSITES + lane * 8);
  float4 d1 = *(const float4*)(row + NSITES + lane * 8 + 4);

  unsigned mu =
      (u0.x > 0.5f ? 1u : 0u)   | (u0.y > 0.5f ? 2u : 0u)   |
      (u0.z > 0.5f ? 4u : 0u)   | (u0.w > 0.5f ? 8u : 0u)   |
      (u1.x > 0.5f ? 16u : 0u)  | (u1.y > 0.5f ? 32u : 0u)  |
      (u1.z > 0.5f ? 64u : 0u)  | (u1.w > 0.5f ? 128u : 0u);
  unsigned md =
      (d0.x > 0.5f ? 1u : 0u)   | (d0.y > 0.5f ? 2u : 0u)   |
      (d0.z > 0.5f ? 4u : 0u)   | (d0.w > 0.5f ? 8u : 0u)   |
      (d1.x > 0.5f ? 16u : 0u)  | (d1.y > 0.5f ? 32u : 0u)  |
      (d1.z > 0.5f ? 64u : 0u)  | (d1.w > 0.5f ? 128u : 0u);

  // ---- exclusive prefix scan of set-bit counts (up | down packed) --------
  int cnt = __popc(mu) | (__popc(md) << 16);
  int inc = cnt;
#pragma unroll
  for (int off = 1; off < 32; off <<= 1) {
    int t = bperm_i(lane - off, inc);
    inc += (lane >= off) ? t : 0;
  }
  const int ex = inc - cnt;
  const int baseU = ex & 0xFFFF;
  const int baseD = ex >> 16;

  // ---- scatter sorted indices (rank order == ascending position) --------
  {
    unsigned m = mu; int r = baseU;
    while (m) { int p = __builtin_ctz(m); m &= m - 1; idxU[wid][r++] = lane * 8 + p; }
  }
  {
    unsigned m = md; int r = baseD;
    while (m) { int p = __builtin_ctz(m); m &= m - 1; idxD[wid][r++] = lane * 8 + p; }
  }
  __builtin_amdgcn_wave_barrier();
  asm volatile("s_wait_dscnt 0" ::: "memory");
  __builtin_amdgcn_wave_barrier();

  const int myIdxU = idxU[wid][lane & 15];
  const int myIdxD = idxD[wid][lane & 15];

  // ---- uniform 64-bit masks of occupied K-chunks (4 columns each) -------
  unsigned b0u = __builtin_amdgcn_ballot_w32((mu & 0x0Fu) != 0);
  unsigned b1u = __builtin_amdgcn_ballot_w32((mu & 0xF0u) != 0);
  unsigned b0d = __builtin_amdgcn_ballot_w32((md & 0x0Fu) != 0);
  unsigned b1d = __builtin_amdgcn_ballot_w32((md & 0xF0u) != 0);
  unsigned long long cmU = spread1(b0u) | (spread1(b1u) << 1);
  unsigned long long cmD = spread1(b0d) | (spread1(b1d) << 1);

  // ---- WMMA gather: mats = P * states (exact in f32) --------------------
  const int khalf = (lane >> 4) << 1;  // 0 (lanes 0-15) or 2 (lanes 16-31)
  const int col   = lane & 15;
  v8f accU = gather_wmma(cmU, myIdxU, sU, khalf, col);
  v8f accD = gather_wmma(cmD, myIdxD, sD, khalf, col);

  // ---- transpose C layout -> rows-in-lanes via padded LDS tile ----------
  const int rbase = (lane >> 4) << 3;  // rows 0-7 (lo lanes) / 8-15 (hi lanes)
#pragma unroll
  for (int g = 0; g < 8; ++g) {
    tU[wid][(rbase + g) * TSTRIDE + col] = accU[g];
    tD[wid][(rbase + g) * TSTRIDE + col] = accD[g];
  }
  __builtin_amdgcn_wave_barrier();
  asm volatile("s_wait_dscnt 0" ::: "memory");
  __builtin_amdgcn_wave_barrier();

  // lanes 0-15: up rows; lanes 16-31: down rows — both eliminated in parallel
  const float* myTile = (lane < 16) ? &tU[wid][0] : &tD[wid][0];
  float r[16];
#pragma unroll
  for (int j = 0; j < 16; ++j) r[j] = myTile[(lane & 15) * TSTRIDE + j];

  // ---- Gaussian elimination, pivot-row broadcast via ds_bpermute --------
  float det = 1.0f;
  const int hb = lane & 16;
  const int rowId = lane & 15;
#pragma unroll
  for (int k = 0; k < 16; ++k) {
    float pk[16];
#pragma unroll
    for (int j = 0; j < 16; ++j) {
      if (j >= k) pk[j] = bperm_f(hb | k, r[j]);
    }
    const float akk = pk[k];
    det *= akk;                                      // uniform within half-wave
    const float f = (rowId > k) ? (r[k] * __builtin_amdgcn_rcpf(akk)) : 0.0f;
#pragma unroll
    for (int j = 0; j < 16; ++j) {
      if (j >= k) r[j] -= f * pk[j];
    }
  }

  // det(up) lives in lanes 0-15, det(down) in lanes 16-31
  const float detD = __int_as_float(__builtin_amdgcn_readlane(__float_as_int(det), 16));
  if (lane == 0) out[b] = det * detD;
}

extern "C" void kernel_launch(void* const* d_in, const int* in_sizes, int n_in,
                              void* d_out, int out_size, void* d_ws, size_t ws_size,
                              hipStream_t stream) {
  const float* cfg = (const float*)d_in[0];   // [B, 512] float32
  const float* su  = (const float*)d_in[1];   // [256, 16] float32
  const float* sd  = (const float*)d_in[2];   // [256, 16] float32
  float* out = (float*)d_out;                 // [B] float32

  const int nB = in_sizes[0] / (2 * NSITES);
  const int blocks = (nB + WPB - 1) / WPB;
  slater_det_kernel<<<blocks, 256, 0, stream>>>(cfg, su, sd, out, nB);
}